// LEQL_21492016349591
// MI455X (gfx1250) — compile-verified
//
#include <hip/hip_runtime.h>
#include <math.h>

// ---------------------------------------------------------------------------
// LEQL second-derivative kernel for MI455X (gfx1250).
//
// Roofline: ~28 GFLOP fp32 vs ~2.6 MB of HBM traffic (weights L2-resident,
// 192MB L2) -> compute bound by >1000x. The two dominant per-sample GEMMs
// (512x384 @ 384x16 and 384x512 @ 512x16, N=16 tangent directions) map
// exactly onto V_WMMA_F32_16X16X4_F32 tiles (fp32 in/out keeps Hessian
// fidelity). Everything else is small VALU/LDS work.
// ---------------------------------------------------------------------------

#define NIN   32     // input dim [q;qd]
#define NA    512    // active z rows per layer (z[0:128] of 640 is unused)
#define NH    384    // active h entries per layer (h[0:128]==1 folded to bias)
#define THETA 0.5f
#define EPSR  0.001f

typedef __attribute__((ext_vector_type(2))) float v2f;
typedef __attribute__((ext_vector_type(8))) float v8f;

// workspace layout (float offsets)
#define WS_W0P   0                       // 512*32  packed W0 rows 128:640
#define WS_B0P   (WS_W0P + NA*NIN)       // 512
#define WS_W1P   (WS_B0P + NA)           // 512*384 packed W1[128:640,128:512]
#define WS_W1PT  (WS_W1P + NA*NH)        // 384*512 transpose of W1P
#define WS_B1P   (WS_W1PT + NH*NA)       // 512     b1 + ones-block fold
#define WS_WOP   (WS_B1P + NA)           // 2*384   Wo[:,128:512]
#define WS_BOP   (WS_WOP + 2*NH)         // 2       bo + ones-block fold
#define WS_TOTAL (WS_BOP + 2)            // ~411K floats (~1.65 MB)

__device__ __forceinline__ v8f wmma4(v2f a, v2f b, v8f c) {
  return __builtin_amdgcn_wmma_f32_16x16x4_f32(false, a, false, b, (short)0, c,
                                               false, false);
}

// ---------------------------------------------------------------------------
// Pack active weights (and transpose for the backward GEMM) into workspace.
// ---------------------------------------------------------------------------
__global__ void leql_prep(const float* __restrict__ W0, const float* __restrict__ b0,
                          const float* __restrict__ W1, const float* __restrict__ b1,
                          const float* __restrict__ Wo, const float* __restrict__ bo,
                          float* __restrict__ ws) {
  int idx = blockIdx.x * blockDim.x + threadIdx.x;
  if (idx < NA * NH) {
    int za = idx / NH, ha = idx % NH;
    float w = W1[(128 + za) * 512 + (128 + ha)];
    ws[WS_W1P + za * NH + ha] = w;
    ws[WS_W1PT + ha * NA + za] = w;
  }
  if (idx < NA * NIN) {
    ws[WS_W0P + idx] = W0[128 * NIN + idx];
  }
  if (idx < NA) {
    ws[WS_B0P + idx] = b0[128 + idx];
    float s = b1[128 + idx];
    for (int c = 0; c < 128; ++c) s += W1[(128 + idx) * 512 + c];
    ws[WS_B1P + idx] = s;
  }
  if (idx < 2 * NH) {
    int r = idx / NH, ha = idx % NH;
    ws[WS_WOP + idx] = Wo[r * 512 + 128 + ha];
  }
  if (idx < 2) {
    float s = bo[idx];
    for (int c = 0; c < 128; ++c) s += Wo[idx * 512 + c];
    ws[WS_BOP + idx] = s;
  }
}

// ---------------------------------------------------------------------------
// One block per sample: forward + 16-dir forward tangents + reverse + 16-dir
// tangent-of-reverse (HVPs) + 16x16 Gauss-Jordan solve.
// ---------------------------------------------------------------------------
__global__ __launch_bounds__(256) void leql_main(const float* __restrict__ x,
                                                 const float* __restrict__ ws,
                                                 float* __restrict__ out, int B) {
  const int s    = blockIdx.x;
  const int tid  = threadIdx.x;
  const int lane = tid & 31;
  const int wave = tid >> 5;
  const int half = lane >> 4;   // 0: K0/K1, 1: K2/K3 operand halves
  const int lm   = lane & 15;

  const float* W0p  = ws + WS_W0P;
  const float* b0p  = ws + WS_B0P;
  const float* W1p  = ws + WS_W1P;
  const float* W1pT = ws + WS_W1PT;
  const float* b1p  = ws + WS_B1P;
  const float* Wop  = ws + WS_WOP;
  const float* bop  = ws + WS_BOP;

  __shared__ float sA[NA * 16];   // zdot1 -> zbar1dot -> zbar0dot
  __shared__ float sB[NH * 16];   // hdot0 -> hdot1 -> vdot0
  __shared__ float sZ0[NA], sZ1[NA];
  __shared__ float sH0[NH], sH1[NH];
  __shared__ float sV1[NH], sV0[NH];
  __shared__ float sZb[NA];       // zbar1 then zbar0 (barrier-separated reuse)
  __shared__ float sGd[32 * 16];  // gdot = 32x16 mixed Hessian block
  __shared__ float sU[NIN], sYd[32], sT2[16], sT3[16], sY[2], sJ[16];
  __shared__ float sM[16 * 17];
  __shared__ int   sPiv;

  // ---- stage 1: load u = [q;qd] -------------------------------------------
  if (tid < NIN) sU[tid] = x[(size_t)s * NIN + tid];
  __syncthreads();

  // ---- stage 2: z0 = W0p @ u + b0p ----------------------------------------
  for (int za = tid; za < NA; za += 256) {
    float acc = b0p[za];
    for (int c = 0; c < NIN; ++c) acc += W0p[za * NIN + c] * sU[c];
    sZ0[za] = acc;
  }
  __syncthreads();

  // ---- stage 3+4+5: h0, hdot0 (into sB), z1 -------------------------------
  if (tid < NH) {
    int ha = tid;
    if (ha < 128)      sH0[ha] = __sinf(sZ0[ha]);
    else if (ha < 256) sH0[ha] = __cosf(sZ0[ha]);
    else               sH0[ha] = sZ0[ha] * sZ0[ha + 128];
  }
  if (tid < NH) {   // zdot0(za,d) == W0p[za*32 + 16 + d] (constant tangents)
    int ha = tid;
    if (ha < 128) {
      float cs = __cosf(sZ0[ha]);
      for (int d = 0; d < 16; ++d) sB[ha * 16 + d] = cs * W0p[ha * NIN + 16 + d];
    } else if (ha < 256) {
      float sn = __sinf(sZ0[ha]);
      for (int d = 0; d < 16; ++d) sB[ha * 16 + d] = -sn * W0p[ha * NIN + 16 + d];
    } else {
      int i = ha - 256;
      float a = sZ0[256 + i], b = sZ0[384 + i];
      for (int d = 0; d < 16; ++d)
        sB[ha * 16 + d] = b * W0p[(256 + i) * NIN + 16 + d] +
                          a * W0p[(384 + i) * NIN + 16 + d];
    }
  }
  __syncthreads();
  for (int za = tid; za < NA; za += 256) {
    float acc = b1p[za];
    for (int ha = 0; ha < NH; ++ha) acc += W1p[za * NH + ha] * sH0[ha];
    sZ1[za] = acc;
  }
  __syncthreads();

  // ---- stage 6: GEMM1 (WMMA): sA(512x16) = W1p(512x384) @ sB(384x16) ------
  {
    v8f a0 = {}, a1 = {}, a2 = {}, a3 = {};
    const float* abase = W1p + (wave * 64 + lm) * NH + 2 * half;
    for (int k = 0; k < NH; k += 4) {
      v2f bb;
      bb.x = sB[(k + 2 * half + 0) * 16 + lm];
      bb.y = sB[(k + 2 * half + 1) * 16 + lm];
      const float* ap = abase + k;
      v2f aa;
      aa.x = ap[0];           aa.y = ap[1];            a0 = wmma4(aa, bb, a0);
      aa.x = ap[16 * NH];     aa.y = ap[16 * NH + 1];  a1 = wmma4(aa, bb, a1);
      aa.x = ap[32 * NH];     aa.y = ap[32 * NH + 1];  a2 = wmma4(aa, bb, a2);
      aa.x = ap[48 * NH];     aa.y = ap[48 * NH + 1];  a3 = wmma4(aa, bb, a3);
    }
    int rb = wave * 64 + half * 8;
    for (int v = 0; v < 8; ++v) {
      sA[(rb +      v) * 16 + lm] = a0[v];
      sA[(rb + 16 + v) * 16 + lm] = a1[v];
      sA[(rb + 32 + v) * 16 + lm] = a2[v];
      sA[(rb + 48 + v) * 16 + lm] = a3[v];
    }
  }
  __syncthreads();

  // ---- stage 7+8: h1, hdot1 (sB <- from sA) -------------------------------
  if (tid < NH) {
    int ha = tid;
    if (ha < 128)      sH1[ha] = __sinf(sZ1[ha]);
    else if (ha < 256) sH1[ha] = __cosf(sZ1[ha]);
    else               sH1[ha] = sZ1[ha] * sZ1[ha + 128];
  }
  if (tid < NH) {
    int ha = tid;
    if (ha < 128) {
      float cs = __cosf(sZ1[ha]);
      for (int d = 0; d < 16; ++d) sB[ha * 16 + d] = cs * sA[ha * 16 + d];
    } else if (ha < 256) {
      float sn = __sinf(sZ1[ha]);
      for (int d = 0; d < 16; ++d) sB[ha * 16 + d] = -sn * sA[ha * 16 + d];
    } else {
      int i = ha - 256;
      float a = sZ1[256 + i], b = sZ1[384 + i];
      for (int d = 0; d < 16; ++d)
        sB[ha * 16 + d] = b * sA[(256 + i) * 16 + d] + a * sA[(384 + i) * 16 + d];
    }
  }
  __syncthreads();

  // ---- stage 9: y = Wop@h1 + bop ; ydot = Wop@hdot1 -----------------------
  if (tid < 2) {
    float acc = bop[tid];
    for (int ha = 0; ha < NH; ++ha) acc += Wop[tid * NH + ha] * sH1[ha];
    sY[tid] = acc;
  }
  if (tid < 32) {
    int r = tid >> 4, d = tid & 15;
    float acc = 0.f;
    for (int ha = 0; ha < NH; ++ha) acc += Wop[r * NH + ha] * sB[ha * 16 + d];
    sYd[r * 16 + d] = acc;
  }
  __syncthreads();

  // ---- stage 10+11: adjoint seeds, tangent seeds, v1, outputs lg/y2 -------
  const float y1 = sY[0], y2 = sY[1];
  const bool  fl = (y2 > THETA);
  const float iy2  = fl ? 1.f / y2 : 0.f;
  const float alpha = iy2;                       // d lg / d y1
  const float beta  = fl ? -y1 * iy2 * iy2 : 0.f;  // d lg / d y2
  if (tid == 0) {
    out[(size_t)B * 16 + s] = fl ? y1 * iy2 : 0.f;
    out[(size_t)B * 17 + s] = y2;
  }
  if (tid < 16) {
    float yd1 = sYd[tid], yd2 = sYd[16 + tid];
    sT2[tid] = fl ? -yd2 * iy2 * iy2 : 0.f;                       // alpha_dot
    sT3[tid] = fl ? (-yd1 * iy2 * iy2 + 2.f * y1 * yd2 * iy2 * iy2 * iy2) : 0.f;
  }
  if (tid < NH) sV1[tid] = alpha * Wop[tid] + beta * Wop[NH + tid];
  __syncthreads();

  // ---- stage 12: zbar1 ----------------------------------------------------
  for (int za = tid; za < NA; za += 256) {
    float r;
    if (za < 128)      r =  __cosf(sZ1[za]) * sV1[za];
    else if (za < 256) r = -__sinf(sZ1[za]) * sV1[za];
    else if (za < 384) r = sZ1[za + 128] * sV1[za];            // b * v
    else               r = sZ1[za - 128] * sV1[za - 128];      // a * v
    sZb[za] = r;
  }
  __syncthreads();

  // ---- stage 13: v0 = W1pT @ zbar1 ----------------------------------------
  if (tid < NH) {
    float acc = 0.f;
    for (int za = 0; za < NA; ++za) acc += W1pT[tid * NA + za] * sZb[za];
    sV0[tid] = acc;
  }
  __syncthreads();

  // ---- stage 14: zbar0 (reuse sZb) ----------------------------------------
  for (int za = tid; za < NA; za += 256) {
    float r;
    if (za < 128)      r =  __cosf(sZ0[za]) * sV0[za];
    else if (za < 256) r = -__sinf(sZ0[za]) * sV0[za];
    else if (za < 384) r = sZ0[za + 128] * sV0[za];
    else               r = sZ0[za - 128] * sV0[za - 128];
    sZb[za] = r;
  }
  __syncthreads();

  // ---- stage 15: jcob = (W0p^T zbar0)[0:16] ; stage 16: zbar1dot (in sA) --
  if (tid < 16) {
    float acc = 0.f;
    for (int za = 0; za < NA; ++za) acc += W0p[za * NIN + tid] * sZb[za];
    sJ[tid] = acc;
  }
  if (tid < 256) {   // sin/cos rows of zbar1dot (read+write own sA row)
    int za = tid;
    float z = sZ1[za], v = sV1[za];
    float sn = __sinf(z), cs = __cosf(z);
    float w0j = Wop[za], w1j = Wop[NH + za];
    if (za < 128) {
      for (int d = 0; d < 16; ++d) {
        float vd = sT2[d] * w0j + sT3[d] * w1j;
        sA[za * 16 + d] = -sn * sA[za * 16 + d] * v + cs * vd;
      }
    } else {
      for (int d = 0; d < 16; ++d) {
        float vd = sT2[d] * w0j + sT3[d] * w1j;
        sA[za * 16 + d] = -cs * sA[za * 16 + d] * v - sn * vd;
      }
    }
  }
  if (tid < 128) {   // mult pairs: one thread owns both coupled rows
    int i = tid;
    float v = sV1[256 + i];
    float a = sZ1[256 + i], b = sZ1[384 + i];
    float w0j = Wop[256 + i], w1j = Wop[NH + 256 + i];
    float zda[16], zdb[16];
    for (int d = 0; d < 16; ++d) {
      zda[d] = sA[(256 + i) * 16 + d];
      zdb[d] = sA[(384 + i) * 16 + d];
    }
    for (int d = 0; d < 16; ++d) {
      float vd = sT2[d] * w0j + sT3[d] * w1j;
      sA[(256 + i) * 16 + d] = zdb[d] * v + b * vd;
      sA[(384 + i) * 16 + d] = zda[d] * v + a * vd;
    }
  }
  __syncthreads();

  // ---- stage 17: GEMM2 (WMMA): sB(384x16) = W1pT(384x512) @ sA(512x16) ----
  {
    v8f a0 = {}, a1 = {}, a2 = {};
    const float* abase = W1pT + (wave * 48 + lm) * NA + 2 * half;
    for (int k = 0; k < NA; k += 4) {
      v2f bb;
      bb.x = sA[(k + 2 * half + 0) * 16 + lm];
      bb.y = sA[(k + 2 * half + 1) * 16 + lm];
      const float* ap = abase + k;
      v2f aa;
      aa.x = ap[0];           aa.y = ap[1];            a0 = wmma4(aa, bb, a0);
      aa.x = ap[16 * NA];     aa.y = ap[16 * NA + 1];  a1 = wmma4(aa, bb, a1);
      aa.x = ap[32 * NA];     aa.y = ap[32 * NA + 1];  a2 = wmma4(aa, bb, a2);
    }
    int rb = wave * 48 + half * 8;
    for (int v = 0; v < 8; ++v) {
      sB[(rb +      v) * 16 + lm] = a0[v];
      sB[(rb + 16 + v) * 16 + lm] = a1[v];
      sB[(rb + 32 + v) * 16 + lm] = a2[v];
    }
  }
  __syncthreads();

  // ---- stage 18: zbar0dot into sA (reads sB, sV0, sZ0, const zdot0) -------
  for (int za = tid; za < NA; za += 256) {
    if (za < 128) {
      float sn = __sinf(sZ0[za]), cs = __cosf(sZ0[za]);
      for (int d = 0; d < 16; ++d)
        sA[za * 16 + d] = -sn * W0p[za * NIN + 16 + d] * sV0[za] + cs * sB[za * 16 + d];
    } else if (za < 256) {
      float sn = __sinf(sZ0[za]), cs = __cosf(sZ0[za]);
      for (int d = 0; d < 16; ++d)
        sA[za * 16 + d] = -cs * W0p[za * NIN + 16 + d] * sV0[za] - sn * sB[za * 16 + d];
    } else if (za < 384) {
      int i = za - 256;
      float v = sV0[256 + i], b = sZ0[384 + i];
      for (int d = 0; d < 16; ++d)
        sA[za * 16 + d] = W0p[(384 + i) * NIN + 16 + d] * v + b * sB[(256 + i) * 16 + d];
    } else {
      int i = za - 384;
      float v = sV0[256 + i], a = sZ0[256 + i];
      for (int d = 0; d < 16; ++d)
        sA[za * 16 + d] = W0p[(256 + i) * NIN + 16 + d] * v + a * sB[(256 + i) * 16 + d];
    }
  }
  __syncthreads();

  // ---- stage 19: gdot(32x16) = W0p^T @ zbar0dot ---------------------------
  for (int idx = tid; idx < 32 * 16; idx += 256) {
    int r = idx >> 4, d = idx & 15;
    float acc = 0.f;
    for (int za = 0; za < NA; ++za) acc += W0p[za * NIN + r] * sA[za * 16 + d];
    sGd[idx] = acc;
  }
  __syncthreads();

  // ---- stage 20: assemble M = H_qdqd + eps*I and rhs = jcob - h2@q --------
  if (tid < 256) {
    int i = tid >> 4, j = tid & 15;
    sM[i * 17 + j] = sGd[(16 + i) * 16 + j] + ((i == j) ? EPSR : 0.f);
  }
  if (tid < 16) {
    float acc = sJ[tid];
    for (int d = 0; d < 16; ++d) acc -= sGd[tid * 16 + d] * sU[d];
    sM[tid * 17 + 16] = acc;
  }
  __syncthreads();

  // ---- stage 21: Gauss-Jordan with partial pivoting -----------------------
  for (int k = 0; k < 16; ++k) {
    if (tid == 0) {
      int p = k; float best = fabsf(sM[k * 17 + k]);
      for (int i = k + 1; i < 16; ++i) {
        float vv = fabsf(sM[i * 17 + k]);
        if (vv > best) { best = vv; p = i; }
      }
      sPiv = p;
    }
    __syncthreads();
    if (tid < 17) {
      int p = sPiv;
      float a = sM[k * 17 + tid], b = sM[p * 17 + tid];
      sM[k * 17 + tid] = b;
      sM[p * 17 + tid] = a;
    }
    __syncthreads();
    if (tid < 16 && tid != k) {
      float f = sM[tid * 17 + k] / sM[k * 17 + k];
      for (int j = k; j < 17; ++j) sM[tid * 17 + j] -= f * sM[k * 17 + j];
    }
    __syncthreads();
  }

  // ---- stage 22: qdd ------------------------------------------------------
  if (tid < 16)
    out[(size_t)s * 16 + tid] = sM[tid * 17 + 16] / sM[tid * 17 + tid];
}

extern "C" void kernel_launch(void* const* d_in, const int* in_sizes, int n_in,
                              void* d_out, int out_size, void* d_ws, size_t ws_size,
                              hipStream_t stream) {
  (void)n_in; (void)out_size; (void)ws_size;
  const float* x  = (const float*)d_in[0];
  const float* W0 = (const float*)d_in[1];
  const float* b0 = (const float*)d_in[2];
  const float* W1 = (const float*)d_in[3];
  const float* b1 = (const float*)d_in[4];
  const float* Wo = (const float*)d_in[5];
  const float* bo = (const float*)d_in[6];
  float* ws  = (float*)d_ws;     // needs WS_TOTAL*4 ~ 1.65 MB
  float* out = (float*)d_out;    // [qdd(B*16) | lg(B) | y2(B)]
  const int B = in_sizes[0] / NIN;

  const int prepN = NA * NH;     // largest packing task
  leql_prep<<<(prepN + 255) / 256, 256, 0, stream>>>(W0, b0, W1, b1, Wo, bo, ws);
  leql_main<<<B, 256, 0, stream>>>(x, ws, out, B);
}